// AttODEblock_5394478923813
// MI455X (gfx1250) — compile-verified
//
#include <hip/hip_runtime.h>
#include <math.h>

#define N_NODES 50000
#define N_EDGES 800000
#define DIM     128
#define HEADS   4
#define RSQRT_DK 0.17677669529663687f   // 1/sqrt(32)
#define STEPSZ   0.25f

typedef float v2f __attribute__((ext_vector_type(2)));
typedef float v8f __attribute__((ext_vector_type(8)));

// float atomic max via signed/unsigned int ordering trick
__device__ __forceinline__ float atomic_max_f32(float* addr, float val) {
  if (val >= 0.0f)
    return __int_as_float(atomicMax((int*)addr, __float_as_int(val)));
  return __uint_as_float(atomicMin((unsigned int*)addr, __float_as_uint(val)));
}

// ---- init: ax = 0, segsum = 0, segmax = -inf ----
__global__ __launch_bounds__(256) void k_init(float* __restrict__ segmax,
                                              float* __restrict__ segsum,
                                              float* __restrict__ ax) {
  const int i = blockIdx.x * 256 + threadIdx.x;
  if (i < N_NODES * DIM) ax[i] = 0.0f;
  if (i < N_NODES * HEADS) {
    segmax[i] = __int_as_float(0xFF800000);  // -inf
    segsum[i] = 0.0f;
  }
}

// ---- fused Q/K projection: one wave per 16x16 tile, fp32 WMMA K=4 ----
// A 16x4 f32 layout: lane(0-15) row M=lane, VGPR0/1 = K{0,1}; lanes 16-31 = K{2,3}
// B 4x16 f32 layout: VGPR0 rows K{0|2}, VGPR1 rows K{1|3} by half-wave; N = lane&15
// C/D 16x16 f32: VGPR g = row (g + 8*half), N = lane&15
__global__ __launch_bounds__(32) void k_gemm_qk(
    const float* __restrict__ x,
    const float* __restrict__ Wq, const float* __restrict__ bq,
    const float* __restrict__ Wk, const float* __restrict__ bk,
    float* __restrict__ q, float* __restrict__ k) {
  const int lane = threadIdx.x;
  const int half = lane >> 4;
  const int lid  = lane & 15;
  const int col0 = blockIdx.x * 16;
  const int row0 = blockIdx.y * 16;
  const int which = blockIdx.z;
  const float* __restrict__ W   = which ? Wk : Wq;
  const float* __restrict__ bia = which ? bk : bq;
  float* __restrict__ out       = which ? k  : q;

  v8f c = {};
  const float* ap = x + (size_t)(row0 + lid) * DIM + 2 * half;
  const float* bp = W + (size_t)(2 * half) * DIM + col0 + lid;
#pragma unroll 4
  for (int kk = 0; kk < DIM; kk += 4) {
    v2f a, b;
    a.x = ap[kk];
    a.y = ap[kk + 1];
    b.x = bp[(size_t)kk * DIM];
    b.y = bp[(size_t)(kk + 1) * DIM];
    c = __builtin_amdgcn_wmma_f32_16x16x4_f32(false, a, false, b,
                                              (short)0, c, false, false);
  }
  const float bv = bia[col0 + lid];
  float* op = out + (size_t)(row0 + 8 * half) * DIM + col0 + lid;
#pragma unroll
  for (int g = 0; g < 8; ++g) op[(size_t)g * DIM] = c[g] + bv;
}

// ---- per-edge multi-head scores + segment max (one wave per edge) ----
__global__ __launch_bounds__(256) void k_scores(
    const float* __restrict__ q, const float* __restrict__ k,
    const long long* __restrict__ ei,
    float* __restrict__ scores, float* __restrict__ segmax) {
  const int e = blockIdx.x * 8 + (threadIdx.x >> 5);
  if (e >= N_EDGES) return;
  const int lane = threadIdx.x & 31;
  const int s = (int)ei[e];
  const int d = (int)ei[N_EDGES + e];
  const float4 a = ((const float4*)(q + (size_t)s * DIM))[lane];
  const float4 b = ((const float4*)(k + (size_t)d * DIM))[lane];
  float p = a.x * b.x + a.y * b.y + a.z * b.z + a.w * b.w;
  // heads occupy 8-lane groups (lane*4/32): reduce inside the group
  p += __shfl_xor(p, 1);
  p += __shfl_xor(p, 2);
  p += __shfl_xor(p, 4);
  if ((lane & 7) == 0) {
    const int h = lane >> 3;
    const float sc = p * RSQRT_DK;
    scores[(size_t)e * HEADS + h] = sc;
    atomic_max_f32(&segmax[(size_t)d * HEADS + h], sc);
  }
}

// ---- exp(score - max) and segment sum ----
__global__ __launch_bounds__(256) void k_expsum(
    float* __restrict__ scores, const float* __restrict__ segmax,
    float* __restrict__ segsum, const long long* __restrict__ ei) {
  const int i = blockIdx.x * 256 + threadIdx.x;   // over E*H
  const int e = i >> 2;
  const int h = i & 3;
  const int d = (int)ei[N_EDGES + e];
  const float ex = expf(scores[i] - segmax[d * HEADS + h]);
  scores[i] = ex;
  atomicAdd(&segsum[d * HEADS + h], ex);
}

// ---- head-averaged attention per edge ----
__global__ __launch_bounds__(256) void k_attmean(
    const float* __restrict__ scores, const float* __restrict__ segsum,
    const long long* __restrict__ ei, float* __restrict__ att) {
  const int e = blockIdx.x * 256 + threadIdx.x;   // over E (exact)
  const int d = (int)ei[N_EDGES + e];
  float s = 0.0f;
#pragma unroll
  for (int h = 0; h < HEADS; ++h)
    s += scores[(size_t)e * HEADS + h] / (segsum[d * HEADS + h] + 1e-16f);
  att[e] = 0.25f * s;
}

// ---- Euler scatter: ax[src] += att[e] * state[dst] (one wave per edge) ----
__global__ __launch_bounds__(256) void k_scatter(
    const float* __restrict__ state, const float* __restrict__ att,
    const long long* __restrict__ ei, float* __restrict__ ax) {
  const int e = blockIdx.x * 8 + (threadIdx.x >> 5);
  if (e >= N_EDGES) return;
  const int lane = threadIdx.x & 31;
  const int s = (int)ei[e];
  const int d = (int)ei[N_EDGES + e];
  const float a = att[e];
  const float4 v = ((const float4*)(state + (size_t)d * DIM))[lane];
  float* p = ax + (size_t)s * DIM + lane * 4;
  atomicAdd(p + 0, a * v.x);
  atomicAdd(p + 1, a * v.y);
  atomicAdd(p + 2, a * v.z);
  atomicAdd(p + 3, a * v.w);
}

// ---- Euler update: nxt = cur + h*(ax - cur); re-zero ax for next step ----
__global__ __launch_bounds__(256) void k_update(
    const float* __restrict__ cur, float* __restrict__ ax,
    float* __restrict__ nxt) {
  const int i = blockIdx.x * 256 + threadIdx.x;   // N*D exactly divisible
  const float c0 = cur[i];
  const float a  = ax[i];
  nxt[i] = c0 + STEPSZ * (a - c0);
  ax[i] = 0.0f;
}

extern "C" void kernel_launch(void* const* d_in, const int* in_sizes, int n_in,
                              void* d_out, int out_size, void* d_ws, size_t ws_size,
                              hipStream_t stream) {
  (void)in_sizes; (void)n_in; (void)out_size; (void)ws_size;
  const float*     x  = (const float*)d_in[0];
  const float*     Wq = (const float*)d_in[1];
  const float*     bq = (const float*)d_in[2];
  const float*     Wk = (const float*)d_in[3];
  const float*     bk = (const float*)d_in[4];
  const long long* ei = (const long long*)d_in[5];   // int64 [2, E]

  float* ws     = (float*)d_ws;
  float* q      = ws;                                    // N*D (reused as Euler buf A)
  float* kbuf   = q      + (size_t)N_NODES * DIM;        // N*D (reused as Euler buf B)
  float* scores = kbuf   + (size_t)N_NODES * DIM;        // E*H (raw then exp)
  float* segmax = scores + (size_t)N_EDGES * HEADS;      // N*H
  float* segsum = segmax + (size_t)N_NODES * HEADS;      // N*H
  float* att    = segsum + (size_t)N_NODES * HEADS;      // E
  float* ax     = att    + (size_t)N_EDGES;              // N*D

  k_init<<<(N_NODES * DIM) / 256, 256, 0, stream>>>(segmax, segsum, ax);
  k_gemm_qk<<<dim3(DIM / 16, N_NODES / 16, 2), 32, 0, stream>>>(
      x, Wq, bq, Wk, bk, q, kbuf);
  k_scores<<<N_EDGES / 8, 256, 0, stream>>>(q, kbuf, ei, scores, segmax);
  k_expsum<<<(N_EDGES * HEADS) / 256, 256, 0, stream>>>(scores, segmax, segsum, ei);
  k_attmean<<<N_EDGES / 256, 256, 0, stream>>>(scores, segsum, ei, att);

  // 4 Euler steps; q/k buffers are free now and serve as state ping-pong.
  const float* cur = x;
  for (int st = 0; st < 4; ++st) {
    float* nxt = (st == 3) ? (float*)d_out : ((st & 1) ? kbuf : q);
    k_scatter<<<N_EDGES / 8, 256, 0, stream>>>(cur, att, ei, ax);
    k_update<<<(N_NODES * DIM) / 256, 256, 0, stream>>>(cur, ax, nxt);
    cur = nxt;
  }
}